// ComPosHGNN_73976516706654
// MI455X (gfx1250) — compile-verified
//
#include <hip/hip_runtime.h>
#include <hip/hip_bf16.h>

// ---------------------------------------------------------------------------
// ComPosHGNN: 4-relation heterogeneous GraphConv on MI455X (gfx1250, wave32).
//   per relation r:
//     deg_out = segsum(w, src); deg_in = segsum(w, dst)
//     wn = w * rsqrt(deg_out[src]*deg_in[dst])
//     proj = h_src @ W_r                (fp32 WMMA 16x16x4)
//     agg[dst] += proj[src] * wn        (fp32 global atomics, wave-per-edge)
//     out_half += 0.5 * relu(agg + b_r)
// Scatter phase dominates (~4 GB HBM/L2 traffic) -> needs native
// global_atomic_add_f32, not a CAS loop. scatter_kernel placed first in the
// module so the disasm snippet exposes its atomic codegen.
// ---------------------------------------------------------------------------

#define NN    40000
#define DD    128
#define EE    1000000

typedef float v2f __attribute__((ext_vector_type(2)));
typedef float v8f __attribute__((ext_vector_type(8)));

__device__ __forceinline__ void atomAddF(float* p, float v) {
  __hip_atomic_fetch_add(p, v, __ATOMIC_RELAXED, __HIP_MEMORY_SCOPE_AGENT);
}

// ---- edge scatter: agg[dst] += proj[src] * wn ---------------------------
// One wave per edge; each lane moves a float4 (32 lanes x 4 = 128 floats).
__global__ void scatter_kernel(const int* __restrict__ src,
                               const int* __restrict__ dst,
                               const float* __restrict__ w,
                               const float* __restrict__ deg_out,
                               const float* __restrict__ deg_in,
                               const float* __restrict__ proj,
                               float* __restrict__ agg,
                               int n_edges) {
  const int lane = threadIdx.x & 31;
  const int edge = blockIdx.x * (blockDim.x >> 5) + (threadIdx.x >> 5);
  if (edge >= n_edges) return;

  const int   s  = src[edge];
  const int   d  = dst[edge];
  const float wn = w[edge] * rsqrtf(deg_out[s] * deg_in[d]);

  const float4 v = ((const float4*)(proj + (size_t)s * DD))[lane];
  float* ag = agg + (size_t)d * DD + lane * 4;
  atomAddF(ag + 0, v.x * wn);
  atomAddF(ag + 1, v.y * wn);
  atomAddF(ag + 2, v.z * wn);
  atomAddF(ag + 3, v.w * wn);
}

// ---- degree pass: weighted out/in degrees -------------------------------
__global__ void deg_kernel(const int* __restrict__ src,
                           const int* __restrict__ dst,
                           const float* __restrict__ w,
                           float* __restrict__ deg_out,
                           float* __restrict__ deg_in,
                           int n_edges) {
  int e = blockIdx.x * blockDim.x + threadIdx.x;
  if (e < n_edges) {
    float wv = w[e];
    atomAddF(&deg_out[src[e]], wv);
    atomAddF(&deg_in[dst[e]], wv);
  }
}

// ---- projection: P = H @ W via V_WMMA_F32_16X16X4_F32 -------------------
// One wave computes one 16x16 tile of P. K=128 in 32 steps of 4.
// A (16x4 f32):  lane m = l%16, VGPR v, half = l/16  -> H[m0+m][k + 2*half + v]
// B (4x16 f32):  lane n = l%16, VGPR v, half = l/16  -> W[k + 2*half + v][n0+n]
// C/D (16x16):   VGPR v, lane l -> P[m0 + v + 8*(l/16)][n0 + l%16]
__global__ void gemm_wmma_f32(const float* __restrict__ H,
                              const float* __restrict__ W,
                              float* __restrict__ P,
                              int nrows) {
  const int lane = threadIdx.x;                       // 0..31
  const int m0   = blockIdx.x * 16;
  const int n0   = (blockIdx.y * blockDim.y + threadIdx.y) * 16;
  const int mL   = lane & 15;
  const int half = lane >> 4;
  if (m0 >= nrows) return;                            // wave-uniform guard

  const float* hrow = H + (size_t)(m0 + mL) * DD;
  v8f acc = {0.f, 0.f, 0.f, 0.f, 0.f, 0.f, 0.f, 0.f};

#pragma unroll 4
  for (int k = 0; k < DD; k += 4) {
    const int ka = k + 2 * half;                      // even -> 8B aligned
    v2f a = *(const v2f*)(hrow + ka);                 // A fragment (2 VGPRs)
    v2f b;
    b.x = W[(size_t)ka * DD + n0 + mL];               // B fragment (2 VGPRs)
    b.y = W[(size_t)(ka + 1) * DD + n0 + mL];
    acc = __builtin_amdgcn_wmma_f32_16x16x4_f32(
        /*neg_a=*/false, a, /*neg_b=*/false, b,
        /*c_mod=*/(short)0, acc, /*reuse_a=*/false, /*reuse_b=*/false);
  }

#pragma unroll
  for (int v = 0; v < 8; ++v) {
    P[(size_t)(m0 + v + 8 * half) * DD + n0 + mL] = acc[v];
  }
}

// ---- epilogue: out += 0.5 * relu(agg + b), float4 vectorized ------------
__global__ void finalize_kernel(const float4* __restrict__ agg,
                                const float4* __restrict__ bias,
                                float4* __restrict__ out,
                                int n_vec4) {
  int i = blockIdx.x * blockDim.x + threadIdx.x;
  if (i < n_vec4) {
    float4 a = agg[i];
    float4 b = bias[i & (DD / 4 - 1)];
    float4 o = out[i];
    o.x += 0.5f * fmaxf(a.x + b.x, 0.f);
    o.y += 0.5f * fmaxf(a.y + b.y, 0.f);
    o.z += 0.5f * fmaxf(a.z + b.z, 0.f);
    o.w += 0.5f * fmaxf(a.w + b.w, 0.f);
    out[i] = o;
  }
}

extern "C" void kernel_launch(void* const* d_in, const int* in_sizes, int n_in,
                              void* d_out, int out_size, void* d_ws, size_t ws_size,
                              hipStream_t stream) {
  (void)in_sizes; (void)n_in; (void)out_size; (void)ws_size;

  const float* com_emb = (const float*)d_in[0];
  const float* pos_emb = (const float*)d_in[1];

  const int*   demand_src = (const int*)d_in[2];
  const int*   demand_dst = (const int*)d_in[3];
  const float* demand_w   = (const float*)d_in[4];
  const int*   supply_src = (const int*)d_in[5];
  const int*   supply_dst = (const int*)d_in[6];
  const float* supply_w   = (const float*)d_in[7];
  const int*   cflow_src  = (const int*)d_in[8];
  const int*   cflow_dst  = (const int*)d_in[9];
  const float* cflow_w    = (const float*)d_in[10];
  const int*   pflow_src  = (const int*)d_in[11];
  const int*   pflow_dst  = (const int*)d_in[12];
  const float* pflow_w    = (const float*)d_in[13];

  const float* W_demand = (const float*)d_in[14];
  const float* b_demand = (const float*)d_in[15];
  const float* W_supply = (const float*)d_in[16];
  const float* b_supply = (const float*)d_in[17];
  const float* W_cflow  = (const float*)d_in[18];
  const float* b_cflow  = (const float*)d_in[19];
  const float* W_pflow  = (const float*)d_in[20];
  const float* b_pflow  = (const float*)d_in[21];

  // workspace layout (floats)
  float* ws      = (float*)d_ws;
  float* proj    = ws;                               // NN*DD
  float* agg     = ws + (size_t)NN * DD;             // NN*DD
  float* deg_out = ws + (size_t)2 * NN * DD;         // NN
  float* deg_in  = deg_out + NN;                     // NN

  float* out_com = (float*)d_out;                    // [NN, DD]
  float* out_pos = out_com + (size_t)NN * DD;        // [NN, DD]

  hipMemsetAsync(d_out, 0, sizeof(float) * 2ull * NN * DD, stream);

  struct Rel {
    const float* h; const int* src; const int* dst; const float* w;
    const float* W; const float* b; int n_src; int n_dst; float* out;
  };
  const Rel rels[4] = {
    { pos_emb, supply_src, supply_dst, supply_w, W_supply, b_supply, NN, NN, out_com },
    { com_emb, cflow_src,  cflow_dst,  cflow_w,  W_cflow,  b_cflow,  NN, NN, out_com },
    { com_emb, demand_src, demand_dst, demand_w, W_demand, b_demand, NN, NN, out_pos },
    { pos_emb, pflow_src,  pflow_dst,  pflow_w,  W_pflow,  b_pflow,  NN, NN, out_pos },
  };

  for (int r = 0; r < 4; ++r) {
    const Rel& R = rels[r];
    hipMemsetAsync(deg_out, 0, sizeof(float) * R.n_src, stream);
    hipMemsetAsync(deg_in,  0, sizeof(float) * R.n_dst, stream);
    hipMemsetAsync(agg,     0, sizeof(float) * (size_t)R.n_dst * DD, stream);

    deg_kernel<<<(EE + 255) / 256, 256, 0, stream>>>(
        R.src, R.dst, R.w, deg_out, deg_in, EE);

    // 2500 row tiles x 8 col tiles; 4 waves/block each owning one col tile.
    dim3 ggrid(R.n_src / 16, (DD / 16) / 4);
    dim3 gblk(32, 4);
    gemm_wmma_f32<<<ggrid, gblk, 0, stream>>>(R.h, R.W, proj, R.n_src);

    // 8 edges per 256-thread block (wave per edge).
    scatter_kernel<<<(EE + 7) / 8, 256, 0, stream>>>(
        R.src, R.dst, R.w, deg_out, deg_in, proj, agg, EE);

    const int nv4 = R.n_dst * DD / 4;
    finalize_kernel<<<(nv4 + 255) / 256, 256, 0, stream>>>(
        (const float4*)agg, (const float4*)R.b, (float4*)R.out, nv4);
  }
}